// GRAND_13288628814644
// MI455X (gfx1250) — compile-verified
//
#include <hip/hip_runtime.h>
#include <hip/hip_bf16.h>
#include <math.h>

// ---------------- problem constants (match reference) ----------------
#define N_NODES 50000
#define N_EDGES 500000
#define IN_C    128
#define HID_C   256
#define OUT_C   64
#define ORDER   5

typedef __attribute__((ext_vector_type(2))) float v2f;
typedef __attribute__((ext_vector_type(8))) float v8f;

// ---------------- elementwise kernels ----------------

// y = x*0.5 ; hA = x*0.5   (float4 streaming)
__global__ void grand_scale_init(const float* __restrict__ x,
                                 float* __restrict__ y,
                                 float* __restrict__ hA,
                                 int n4) {
    int t = blockIdx.x * blockDim.x + threadIdx.x;
    if (t >= n4) return;
    float4 v = ((const float4*)x)[t];
    v.x *= 0.5f; v.y *= 0.5f; v.z *= 0.5f; v.w *= 0.5f;
    ((float4*)y)[t]  = v;
    ((float4*)hA)[t] = v;
}

// acc += h   (float4 streaming)
__global__ void grand_accum(float* __restrict__ acc,
                            const float* __restrict__ h,
                            int n4) {
    int t = blockIdx.x * blockDim.x + threadIdx.x;
    if (t >= n4) return;
    float4 a = ((float4*)acc)[t];
    float4 b = ((const float4*)h)[t];
    a.x += b.x; a.y += b.y; a.z += b.z; a.w += b.w;
    ((float4*)acc)[t] = a;
}

// out[src[e]] += h[dst[e]] : 32 threads per edge, each thread does 4 floats.
__global__ void grand_spmm_scatter(const int* __restrict__ src,
                                   const int* __restrict__ dst,
                                   const float* __restrict__ h,
                                   float* __restrict__ out) {
    int t = blockIdx.x * blockDim.x + threadIdx.x;      // < N_EDGES*32
    int e = t >> 5;
    if (e >= N_EDGES) return;
    int g = (t & 31) << 2;                              // feature group of 4
    int s = src[e];
    int d = dst[e];
    float4 v = *(const float4*)(h + (size_t)d * IN_C + g);
    float* op = out + (size_t)s * IN_C + g;
    atomicAdd(op + 0, v.x);
    atomicAdd(op + 1, v.y);
    atomicAdd(op + 2, v.z);
    atomicAdd(op + 3, v.w);
}

// ---------------- WMMA f32 GEMM: C = act(A[M,K] @ B[K,N] + bias) ----------------
// One wave computes one 16x16 output tile with V_WMMA_F32_16X16X4_F32.
// A 16-row tile is staged in LDS (stride K+4 to avoid bank conflicts) and
// shared by all waves in the block; each wave owns col-tile (blockIdx.y*waves + wave).
template<int K, int N, bool RELU>
__global__ void wmma_gemm_f32(const float* __restrict__ A,
                              const float* __restrict__ B,
                              const float* __restrict__ bias,
                              float* __restrict__ C,
                              float a_scale) {
    constexpr int KP = K + 4;
    __shared__ float As[16 * KP];

    const int lane   = threadIdx.x & 31;
    const int wave   = threadIdx.x >> 5;
    const int nwaves = blockDim.x >> 5;
    const int row0   = blockIdx.x * 16;

    // cooperative load of the 16xK A tile (coalesced float4)
    for (int i = threadIdx.x * 4; i < 16 * K; i += blockDim.x * 4) {
        int r = i / K;
        int c = i % K;
        float4 v = *(const float4*)(A + (size_t)(row0 + r) * K + c);
        v.x *= a_scale; v.y *= a_scale; v.z *= a_scale; v.w *= a_scale;
        *(float4*)(&As[r * KP + c]) = v;
    }
    __syncthreads();

    const int n0 = (blockIdx.y * nwaves + wave) * 16;   // this wave's col tile
    const int m  = lane & 15;                           // row (A) / col (B) within tile
    const int kb = (lane >> 4) << 1;                    // K sub-offset: 0 or 2

    v8f acc = {};
#pragma unroll 4
    for (int k0 = 0; k0 < K; k0 += 4) {
        v2f a, b;
        a.x = As[m * KP + k0 + kb];
        a.y = As[m * KP + k0 + kb + 1];
        b.x = B[(size_t)(k0 + kb)     * N + n0 + m];
        b.y = B[(size_t)(k0 + kb + 1) * N + n0 + m];
        acc = __builtin_amdgcn_wmma_f32_16x16x4_f32(
            /*neg_a=*/false, a, /*neg_b=*/false, b,
            /*c_mod=*/(short)0, acc, /*reuse_a=*/false, /*reuse_b=*/false);
    }

    const float bv    = bias[n0 + m];
    const int   rbase = row0 + ((lane >> 4) << 3);      // lanes 16-31 hold rows 8..15
#pragma unroll
    for (int r = 0; r < 8; ++r) {
        float v = acc[r] + bv;
        if (RELU) v = fmaxf(v, 0.0f);
        C[(size_t)(rbase + r) * N + n0 + m] = v;
    }
}

// ---------------- in-place row log_softmax over OUT_C=64 cols ----------------
__global__ void grand_log_softmax(float* __restrict__ z) {
    int r = blockIdx.x * blockDim.x + threadIdx.x;
    if (r >= N_NODES) return;
    float* p = z + (size_t)r * OUT_C;
    float4 v[OUT_C / 4];
    float mx = -INFINITY;
#pragma unroll
    for (int i = 0; i < OUT_C / 4; ++i) {
        v[i] = ((const float4*)p)[i];
        mx = fmaxf(mx, fmaxf(fmaxf(v[i].x, v[i].y), fmaxf(v[i].z, v[i].w)));
    }
    float s = 0.0f;
#pragma unroll
    for (int i = 0; i < OUT_C / 4; ++i) {
        s += __expf(v[i].x - mx) + __expf(v[i].y - mx) +
             __expf(v[i].z - mx) + __expf(v[i].w - mx);
    }
    float lse = mx + __logf(s);
#pragma unroll
    for (int i = 0; i < OUT_C / 4; ++i) {
        v[i].x -= lse; v[i].y -= lse; v[i].z -= lse; v[i].w -= lse;
        ((float4*)p)[i] = v[i];
    }
}

// ---------------- host orchestration ----------------
extern "C" void kernel_launch(void* const* d_in, const int* in_sizes, int n_in,
                              void* d_out, int out_size, void* d_ws, size_t ws_size,
                              hipStream_t stream) {
    const float* x  = (const float*)d_in[0];
    const int*   ei = (const int*)d_in[1];          // (2, N_EDGES): row0=src, row1=dst
    const float* W1 = (const float*)d_in[2];
    const float* b1 = (const float*)d_in[3];
    const float* W2 = (const float*)d_in[4];
    const float* b2 = (const float*)d_in[5];
    const float* W3 = (const float*)d_in[6];
    const float* b3 = (const float*)d_in[7];
    const int* src = ei;
    const int* dst = ei + N_EDGES;

    // workspace layout (floats):
    //   y  : N_NODES*IN_C
    //   hA : N_NODES*IN_C
    //   hB : N_NODES*IN_C
    //   H1 : N_NODES*HID_C
    //   H2 aliases [hA,hB] (exactly N_NODES*HID_C floats, dead after propagate)
    float* ws = (float*)d_ws;
    const size_t nf = (size_t)N_NODES * IN_C;       // 6.4M floats
    float* y  = ws;
    float* hA = ws + nf;
    float* hB = ws + 2 * nf;
    float* H1 = ws + 3 * nf;
    float* H2 = hA;                                  // N_NODES*HID_C floats

    const int n4 = (int)(nf / 4);
    const int EB = 256;

    // feats = x*0.5 ; y = feats ; hA = feats
    grand_scale_init<<<(n4 + EB - 1) / EB, EB, 0, stream>>>(x, y, hA, n4);

    // y += sum_{i=1..ORDER} A^i feats
    for (int it = 0; it < ORDER; ++it) {
        hipMemsetAsync(hB, 0, nf * sizeof(float), stream);
        int sthreads = N_EDGES * 32;
        grand_spmm_scatter<<<(sthreads + EB - 1) / EB, EB, 0, stream>>>(src, dst, hA, hB);
        grand_accum<<<(n4 + EB - 1) / EB, EB, 0, stream>>>(y, hB, n4);
        float* t = hA; hA = hB; hB = t;
    }

    // MLP: (y/6) @ W1 + b1 -> relu -> @W2 + b2 -> relu -> @W3 + b3
    {
        dim3 grid(N_NODES / 16, HID_C / (8 * 16));   // (3125, 2)
        wmma_gemm_f32<IN_C, HID_C, true><<<grid, 256, 0, stream>>>(
            y, W1, b1, H1, 1.0f / (ORDER + 1.0f));
    }
    {
        dim3 grid(N_NODES / 16, HID_C / (8 * 16));   // (3125, 2)
        wmma_gemm_f32<HID_C, HID_C, true><<<grid, 256, 0, stream>>>(
            H1, W2, b2, H2, 1.0f);
    }
    {
        dim3 grid(N_NODES / 16, 1);                  // 4 col tiles -> 4 waves
        wmma_gemm_f32<HID_C, OUT_C, false><<<grid, 128, 0, stream>>>(
            H2, W3, b3, (float*)d_out, 1.0f);
    }

    grand_log_softmax<<<(N_NODES + EB - 1) / EB, EB, 0, stream>>>((float*)d_out);
}